// SAGE_31181462569096
// MI455X (gfx1250) — compile-verified
//
#include <hip/hip_runtime.h>
#include <hip/hip_bf16.h>
#include <math.h>

// ---------------------------------------------------------------------------
// GraphSAGE (3 layers) for MI455X / gfx1250, wave32.
//   layer: agg[dst] += w_e * h[src];  out = relu([h || agg] @ W + b)
//   final: log_softmax over 40 classes
// GEMM uses V_WMMA_F32_16X16X4_F32 (exact f32 matrix path) with W pre-packed
// into B-fragment order so the inner loop is branchless and fully coalesced.
// ---------------------------------------------------------------------------

typedef float v2f __attribute__((ext_vector_type(2)));
typedef float v8f __attribute__((ext_vector_type(8)));

// ----------------------------- zero fill -----------------------------------
__global__ void sage_zero(float* __restrict__ p, long long n) {
    long long i = (long long)blockIdx.x * blockDim.x + threadIdx.x;
    if (i < n) p[i] = 0.0f;
}

// --------------------- edge-weighted scatter-add ---------------------------
// One thread handles 4 channels of one edge (float4 gather + 4 f32 atomics).
// h features live in L2 (102MB < 192MB), so gathers/atomics are L2-resident.
__global__ void sage_scatter(const float* __restrict__ H,
                             const int*   __restrict__ ei,
                             const float* __restrict__ ew,
                             float* __restrict__ AGG,
                             int E, int d, int lc /* log2(d/4) */) {
    long long idx = (long long)blockIdx.x * blockDim.x + threadIdx.x;
    long long total = (long long)E << lc;
    if (idx >= total) return;
    int e  = (int)(idx >> lc);
    int c4 = (int)(idx & ((1 << lc) - 1));
    int src = ei[e];
    int dst = ei[E + e];
    float w = ew[e];
    const float4 v = *(const float4*)(H + (size_t)src * d + c4 * 4);
    float* out = AGG + (size_t)dst * d + c4 * 4;
    atomicAdd(out + 0, v.x * w);
    atomicAdd(out + 1, v.y * w);
    atomicAdd(out + 2, v.z * w);
    atomicAdd(out + 3, v.w * w);
}

// --------------------- pack W into B-fragment layout -----------------------
// WB[((kb*nct + ct)*32 + lane)] = { W[kb*4+ksel][col], W[kb*4+ksel+1][col] }
//   ksel = (lane>>4)*2, col = ct*16 + (lane&15); zero-padded for col >= dout.
// A wave reading 32 consecutive float2s gets exactly its 4x16 B fragment.
__global__ void sage_pack_w(const float* __restrict__ W, float2* __restrict__ WB,
                            int K, int dout, int nct) {
    int idx = blockIdx.x * blockDim.x + threadIdx.x;
    int total = (K / 4) * nct * 32;
    if (idx >= total) return;
    int lane = idx & 31;
    int t    = idx >> 5;
    int ct   = t % nct;
    int kb   = t / nct;
    int ksel = (lane >> 4) * 2;
    int col  = ct * 16 + (lane & 15);
    int k    = kb * 4 + ksel;
    float2 v;
    v.x = (col < dout) ? W[(size_t)k       * dout + col] : 0.0f;
    v.y = (col < dout) ? W[(size_t)(k + 1) * dout + col] : 0.0f;
    WB[idx] = v;
}

// ------------------------- WMMA GEMM + bias + relu -------------------------
// OUT[N,dout] = relu?([H || AGG][N,K] @ W[K,dout] + b), W given packed (WB).
// Block: 256 thr = 8 waves. M tile = 32 rows/block (N%32==0).
// K staged through LDS in 256-float chunks; wave = one 16-col tile, two
// 16-row subtiles (two v8f accumulators, 2 WMMAs per k-step of 4).
#define KSTAGE 256
#define KSP    258   // +2 pad: row stride 258 mod 64 banks = 2 -> conflict-free

__global__ __launch_bounds__(256)
void sage_gemm(const float* __restrict__ H, const float* __restrict__ AGG,
               const float2* __restrict__ WB, const float* __restrict__ bias,
               float* __restrict__ OUT,
               int d, int K, int dout, int nct, int relu) {
    extern __shared__ float lds[];             // 32 * KSP floats = 33 KB

    const int tid  = threadIdx.x;
    const int wave = tid >> 5;
    const int lane = tid & 31;
    const int m0   = blockIdx.x * 32;
    const int ct   = blockIdx.y * 8 + wave;    // 16-col tile index
    const int colBase = ct * 16;
    const int col  = colBase + (lane & 15);
    const int ksel = (lane >> 4) * 2;          // A/B lane-half K offset (ISA layout)
    const int r0   = (lane & 15);

    v8f acc0 = {0.f,0.f,0.f,0.f,0.f,0.f,0.f,0.f};
    v8f acc1 = {0.f,0.f,0.f,0.f,0.f,0.f,0.f,0.f};

    const int nStages = K / KSTAGE;            // K in {256, 512}
    for (int s = 0; s < nStages; ++s) {
        const int gBase = s * KSTAGE;
        // Cooperative load: 32 rows x 256 cols, float4-coalesced, concat on the fly.
        for (int i = tid; i < 32 * (KSTAGE / 4); i += 256) {
            int row  = i >> 6;                 // / (KSTAGE/4)
            int c4   = i & 63;
            int gcol = gBase + c4 * 4;
            int node = m0 + row;
            float4 v;
            if (gcol < d) v = *(const float4*)(H   + (size_t)node * d + gcol);
            else          v = *(const float4*)(AGG + (size_t)node * d + (gcol - d));
            float* dp = &lds[row * KSP + c4 * 4];
            dp[0] = v.x; dp[1] = v.y; dp[2] = v.z; dp[3] = v.w;
        }
        __syncthreads();

        if (ct < nct) {
            const float* a0p = &lds[ r0       * KSP + ksel];
            const float* a1p = &lds[(r0 + 16) * KSP + ksel];
            const float2* wb = WB + ((size_t)(gBase >> 2) * nct + ct) * 32 + lane;
            const int wstep  = nct * 32;       // float2s per k-block
#pragma unroll 4
            for (int k0 = 0; k0 < KSTAGE; k0 += 4) {
                v2f a0 = *(const v2f*)(a0p + k0);
                v2f a1 = *(const v2f*)(a1p + k0);
                v2f b  = *(const v2f*)wb;
                wb += wstep;
                acc0 = __builtin_amdgcn_wmma_f32_16x16x4_f32(
                           false, a0, false, b, (short)0, acc0, false, false);
                acc1 = __builtin_amdgcn_wmma_f32_16x16x4_f32(
                           false, a1, false, b, (short)0, acc1, false, false);
            }
        }
        __syncthreads();
    }

    if (ct < nct && col < dout) {
        const float bv = bias[col];
        const int mo = (lane >> 4) * 8;        // C/D layout: lanes16-31 hold M=r+8
#pragma unroll
        for (int r = 0; r < 8; ++r) {
            int m  = m0 + r + mo;
            float v = acc0[r] + bv;
            if (relu) v = fmaxf(v, 0.0f);
            OUT[(size_t)m * dout + col] = v;
            float v2 = acc1[r] + bv;
            if (relu) v2 = fmaxf(v2, 0.0f);
            OUT[(size_t)(m + 16) * dout + col] = v2;
        }
    }
}

// --------------------------- log_softmax (in place) ------------------------
// One wave per node; C = 40 classes (lane covers col and col+32).
__global__ void sage_log_softmax(float* __restrict__ Z, int C) {
    const int node = blockIdx.x;
    const int lane = threadIdx.x;              // blockDim.x == 32
    float* row = Z + (size_t)node * C;
    float v0 = row[lane];                      // lane < 32 <= C
    const int c1 = lane + 32;
    float v1 = (c1 < C) ? row[c1] : -INFINITY;
    float m = fmaxf(v0, v1);
#pragma unroll
    for (int o = 16; o > 0; o >>= 1) m = fmaxf(m, __shfl_xor(m, o, 32));
    float s = expf(v0 - m) + ((c1 < C) ? expf(v1 - m) : 0.0f);
#pragma unroll
    for (int o = 16; o > 0; o >>= 1) s += __shfl_xor(s, o, 32);
    const float ls = logf(s);
    row[lane] = v0 - m - ls;
    if (c1 < C) row[c1] = v1 - m - ls;
}

// ---------------------------------------------------------------------------
extern "C" void kernel_launch(void* const* d_in, const int* in_sizes, int n_in,
                              void* d_out, int out_size, void* d_ws, size_t ws_size,
                              hipStream_t stream) {
    const float* x  = (const float*)d_in[0];
    const int*   ei = (const int*)  d_in[1];
    const float* ew = (const float*)d_in[2];
    const float* W1 = (const float*)d_in[3];
    const float* b1 = (const float*)d_in[4];
    const float* W2 = (const float*)d_in[5];
    const float* b2 = (const float*)d_in[6];
    const float* W3 = (const float*)d_in[7];
    const float* b3 = (const float*)d_in[8];

    const int IN  = 128;
    const int HID = 256;
    const int N   = in_sizes[0] / IN;          // 100000
    const int E   = in_sizes[2];               // 1600000
    const int C   = out_size / N;              // 40

    // col-tile counts (dout padded to 16)
    const int nct1 = HID / 16;                 // 16
    const int nct2 = HID / 16;                 // 16
    const int nct3 = (C + 15) / 16;            // 3

    float*  agg = (float*)d_ws;                         // N*HID
    float*  h1  = agg + (size_t)N * HID;                // N*HID
    float*  h2  = h1  + (size_t)N * HID;                // N*HID
    float2* wb1 = (float2*)(h2 + (size_t)N * HID);      // (2*IN/4)*nct1*32
    float2* wb2 = wb1 + (size_t)(2 * IN  / 4) * nct1 * 32;
    float2* wb3 = wb2 + (size_t)(2 * HID / 4) * nct2 * 32;
    float*  out = (float*)d_out;

    const dim3 blk(256);
    const size_t smem = 32 * KSP * sizeof(float);

    // -------- pack all three weight matrices (tiny) --------
    {
        int t1 = (2 * IN  / 4) * nct1 * 32;
        int t2 = (2 * HID / 4) * nct2 * 32;
        int t3 = (2 * HID / 4) * nct3 * 32;
        sage_pack_w<<<dim3((t1 + 255) / 256), blk, 0, stream>>>(W1, wb1, 2 * IN,  HID, nct1);
        sage_pack_w<<<dim3((t2 + 255) / 256), blk, 0, stream>>>(W2, wb2, 2 * HID, HID, nct2);
        sage_pack_w<<<dim3((t3 + 255) / 256), blk, 0, stream>>>(W3, wb3, 2 * HID, C,   nct3);
    }

    // -------- Layer 1: d=128, K=256, dout=256, relu --------
    {
        long long nz = (long long)N * IN;
        sage_zero<<<dim3((unsigned)((nz + 255) / 256)), blk, 0, stream>>>(agg, nz);
        long long t = (long long)E * (IN / 4);
        sage_scatter<<<dim3((unsigned)((t + 255) / 256)), blk, 0, stream>>>(
            x, ei, ew, agg, E, IN, 5);
        sage_gemm<<<dim3(N / 32, (nct1 + 7) / 8), blk, smem, stream>>>(
            x, agg, wb1, b1, h1, IN, 2 * IN, HID, nct1, 1);
    }
    // -------- Layer 2: d=256, K=512, dout=256, relu --------
    {
        long long nz = (long long)N * HID;
        sage_zero<<<dim3((unsigned)((nz + 255) / 256)), blk, 0, stream>>>(agg, nz);
        long long t = (long long)E * (HID / 4);
        sage_scatter<<<dim3((unsigned)((t + 255) / 256)), blk, 0, stream>>>(
            h1, ei, ew, agg, E, HID, 6);
        sage_gemm<<<dim3(N / 32, (nct2 + 7) / 8), blk, smem, stream>>>(
            h1, agg, wb2, b2, h2, HID, 2 * HID, HID, nct2, 1);
    }
    // -------- Layer 3: d=256, K=512, dout=40, no relu ------
    {
        long long nz = (long long)N * HID;
        sage_zero<<<dim3((unsigned)((nz + 255) / 256)), blk, 0, stream>>>(agg, nz);
        long long t = (long long)E * (HID / 4);
        sage_scatter<<<dim3((unsigned)((t + 255) / 256)), blk, 0, stream>>>(
            h2, ei, ew, agg, E, HID, 6);
        sage_gemm<<<dim3(N / 32, (nct3 + 7) / 8), blk, smem, stream>>>(
            h2, agg, wb3, b3, out, HID, 2 * HID, C, nct3, 0);
    }
    // -------- log_softmax over 40 classes, in place --------
    sage_log_softmax<<<dim3(N), dim3(32), 0, stream>>>(out, C);
}